// TFGridNet_57037165690890
// MI455X (gfx1250) — compile-verified
//
#include <hip/hip_runtime.h>
#include <math.h>

#define B_  2
#define T_  192
#define QF  65
#define C_  48
#define H_  96
#define E_  8
#define VD  12
#define L_  100
#define KS  4
#define QD  16
#define N_  (B_*T_)        // 384
#define BQ  (B_*QF)        // 130
#define BH  (B_*4)         // 8 heads total
#define DQK (QF*E_)        // 520
#define DV  (QF*VD)        // 780
#define DP  (QF*C_)        // 3120
#define LM1 (L_-1)         // 99
#define EPSF 1e-5f

typedef __attribute__((ext_vector_type(2))) float v2f;
typedef __attribute__((ext_vector_type(8))) float v8f;

__device__ __forceinline__ float sigm(float x){ return 1.f/(1.f+__expf(-x)); }

static __device__ __forceinline__ v8f wmma_f32(v2f a, v2f b, v8f c){
  // D = A(16x4,f32) * B(4x16,f32) + C(16x16,f32)
  return __builtin_amdgcn_wmma_f32_16x16x4_f32(false, a, false, b, (short)0, c, false, false);
}

// CDNA5 async global->LDS copy (ASYNCcnt-tracked), 16B per lane.
__device__ __forceinline__ void async_b128(unsigned ldsoff, const float* gsrc){
  asm volatile("global_load_async_to_lds_b128 %0, %1, off" :: "v"(ldsoff), "v"(gsrc) : "memory");
}
__device__ __forceinline__ void wait_async0(){
  asm volatile("s_wait_asynccnt 0x0" ::: "memory");
}

// ---------------- 1. conv(stride ks) + PReLU + LN over C ----------------
__global__ void k_conv_ln(const float* __restrict__ x, const float* __restrict__ cw,
                          const float* __restrict__ cb, const float* __restrict__ act_a,
                          const float* __restrict__ ng, const float* __restrict__ nb,
                          float* __restrict__ u)
{
  int n  = blockIdx.x >> 4;
  int qd = blockIdx.x & 15;
  int co = threadIdx.x;
  __shared__ float vals[C_];
  __shared__ float mv[2];
  if (co < C_) {
    float s = cb[co];
    const float* xp = x + n*DP + (qd*KS)*C_;
    const float* wp = cw + co*C_*KS;
    for (int ci = 0; ci < C_; ++ci)
      for (int k = 0; k < KS; ++k)
        s += xp[k*C_ + ci] * wp[ci*KS + k];
    float a = act_a[0];
    s = (s >= 0.f) ? s : a*s;
    vals[co] = s;
  }
  __syncthreads();
  if (threadIdx.x == 0) {
    float m = 0.f, v = 0.f;
    for (int i = 0; i < C_; ++i) m += vals[i];
    m /= (float)C_;
    for (int i = 0; i < C_; ++i) { float d = vals[i]-m; v += d*d; }
    v /= (float)C_;
    mv[0] = m; mv[1] = rsqrtf(v + EPSF);
  }
  __syncthreads();
  if (co < C_)
    u[(n*QD + qd)*C_ + co] = (vals[co]-mv[0])*mv[1]*ng[co] + nb[co];
}

// ---------------- 2. intra bi-LSTM over frequency ----------------
// 12 waves, 2 gate N-tiles each; weight B-fragments live in VGPRs for the
// whole time loop; input tiles double-buffered via async-to-LDS DMA.
__global__ void __launch_bounds__(384)
k_intra_lstm(const float* __restrict__ u,
             const float* __restrict__ wih_f, const float* __restrict__ whh_f,
             const float* __restrict__ bih_f, const float* __restrict__ bhh_f,
             const float* __restrict__ wih_b, const float* __restrict__ whh_b,
             const float* __restrict__ bih_b, const float* __restrict__ bhh_b,
             float* __restrict__ y)
{
  int dir = blockIdx.x / 24;
  int s0  = (blockIdx.x % 24) * 16;
  const float* wih = dir ? wih_b : wih_f;
  const float* whh = dir ? whh_b : whh_f;
  const float* bih = dir ? bih_b : bih_f;
  const float* bhh = dir ? bhh_b : bhh_f;
  __shared__ __align__(16) float hS[16][H_];
  __shared__ float cS[16][H_];
  __shared__ float gS[16][4*H_];
  __shared__ __align__(16) float uS[2][16][C_];
  int tid = threadIdx.x;
  int wave = tid >> 5, lane = tid & 31;
  int row = lane & 15, kg = (lane >> 4)*2, rb = (lane >> 4)*8;

  // stage this wave's weight fragments in registers (loop-invariant)
  int colA = (wave*2)*16 + (lane & 15);
  int colB = colA + 16;
  v2f bw0[12], bw1[12], bh0[24], bh1[24];
#pragma unroll
  for (int kk = 0; kk < 12; ++kk) {
    bw0[kk][0]=wih[colA*C_+kk*4+kg]; bw0[kk][1]=wih[colA*C_+kk*4+kg+1];
    bw1[kk][0]=wih[colB*C_+kk*4+kg]; bw1[kk][1]=wih[colB*C_+kk*4+kg+1];
  }
#pragma unroll
  for (int kk = 0; kk < 24; ++kk) {
    bh0[kk][0]=whh[colA*H_+kk*4+kg]; bh0[kk][1]=whh[colA*H_+kk*4+kg+1];
    bh1[kk][0]=whh[colB*H_+kk*4+kg]; bh1[kk][1]=whh[colB*H_+kk*4+kg+1];
  }
  float bs0 = bih[colA]+bhh[colA];
  float bs1 = bih[colB]+bhh[colB];

  for (int i = tid; i < 16*H_; i += 384) { hS[i/H_][i%H_] = 0.f; cS[i/H_][i%H_] = 0.f; }
  { // prefill step-0 input tile (async DMA to LDS)
    int s = dir ? (QD-1) : 0;
    if (tid < 192) {
      int r = tid/12, cc = (tid%12)*4;
      async_b128((unsigned)(size_t)&uS[0][r][cc], &u[(s0+r)*QD*C_ + s*C_ + cc]);
    }
  }
  wait_async0();
  __syncthreads();

  for (int st = 0; st < QD; ++st) {
    int buf = st & 1;
    if (st+1 < QD) {           // async-prefetch next step's tile
      int sn = dir ? (QD-2-st) : (st+1);
      if (tid < 192) {
        int r = tid/12, cc = (tid%12)*4;
        async_b128((unsigned)(size_t)&uS[buf^1][r][cc], &u[(s0+r)*QD*C_ + sn*C_ + cc]);
      }
    }
    int s = dir ? (QD-1-st) : st;
    v8f acc0, acc1;
#pragma unroll
    for (int r = 0; r < 8; ++r) { acc0[r]=bs0; acc1[r]=bs1; }
#pragma unroll
    for (int kk = 0; kk < 12; ++kk) {
      v2f a; a[0]=uS[buf][row][kk*4+kg]; a[1]=uS[buf][row][kk*4+kg+1];
      acc0 = wmma_f32(a, bw0[kk], acc0);
      acc1 = wmma_f32(a, bw1[kk], acc1);
    }
#pragma unroll
    for (int kk = 0; kk < 24; ++kk) {
      v2f a; a[0]=hS[row][kk*4+kg]; a[1]=hS[row][kk*4+kg+1];
      acc0 = wmma_f32(a, bh0[kk], acc0);
      acc1 = wmma_f32(a, bh1[kk], acc1);
    }
#pragma unroll
    for (int r = 0; r < 8; ++r) { gS[rb+r][colA] = acc0[r]; gS[rb+r][colB] = acc1[r]; }
    __syncthreads();
    for (int i = tid; i < 16*H_; i += 384) {
      int r = i/H_, j = i%H_;
      float ig = sigm(gS[r][j]);
      float fg = sigm(gS[r][H_+j]);
      float gg = tanhf(gS[r][2*H_+j]);
      float og = sigm(gS[r][3*H_+j]);
      float cc = fg*cS[r][j] + ig*gg;
      cS[r][j] = cc;
      float hh = og*tanhf(cc);
      hS[r][j] = hh;
      y[(s0+r)*QD*2*H_ + s*2*H_ + dir*H_ + j] = hh;
    }
    wait_async0();
    __syncthreads();
  }
}

// ---------------- 3. deconv (stride=ks) + bias + residual ----------------
__global__ void k_deconv(const float* __restrict__ y, const float* __restrict__ dw,
                         const float* __restrict__ db, const float* __restrict__ x,
                         float* __restrict__ intra)
{
  int n = blockIdx.x / QF, qp = blockIdx.x % QF;
  int co = threadIdx.x;
  if (co >= C_) return;
  float s = db[co];
  if (qp < QD*KS) {
    int t = qp >> 2, k = qp & 3;
    const float* yp = y + n*QD*2*H_ + t*2*H_;
    for (int ci = 0; ci < 2*H_; ++ci)
      s += yp[ci] * dw[ci*C_*KS + co*KS + k];
  }
  int o = n*DP + qp*C_ + co;
  intra[o] = s + x[o];
}

// ---------------- 4. LN over C + transpose to [B*Qf, T, C] ----------------
__global__ void k_ln_inter(const float* __restrict__ intra, const float* __restrict__ g,
                           const float* __restrict__ b, float* __restrict__ z)
{
  int n = blockIdx.x / QF, q = blockIdx.x % QF;
  int co = threadIdx.x;
  __shared__ float vals[C_]; __shared__ float mv[2];
  if (co < C_) vals[co] = intra[n*DP + q*C_ + co];
  __syncthreads();
  if (co == 0) {
    float m = 0.f, v = 0.f;
    for (int i = 0; i < C_; ++i) m += vals[i];
    m /= (float)C_;
    for (int i = 0; i < C_; ++i){ float d = vals[i]-m; v += d*d; }
    v /= (float)C_;
    mv[0]=m; mv[1]=rsqrtf(v+EPSF);
  }
  __syncthreads();
  if (co < C_) {
    int bb = n / T_, t = n % T_;
    z[((bb*QF + q)*T_ + t)*C_ + co] = (vals[co]-mv[0])*mv[1]*g[co] + b[co];
  }
}

// ---------------- 5. inter causal LSTM + fused output linear + residual ----------------
__global__ void __launch_bounds__(384)
k_inter_lstm(const float* __restrict__ z, const float* __restrict__ wih,
             const float* __restrict__ whh, const float* __restrict__ bih,
             const float* __restrict__ bhh, const float* __restrict__ h0,
             const float* __restrict__ c0, const float* __restrict__ lw,
             const float* __restrict__ lb, const float* __restrict__ intra,
             float* __restrict__ inter)
{
  int n0 = blockIdx.x * 16;
  __shared__ __align__(16) float hS[16][H_];
  __shared__ float cS[16][H_];
  __shared__ float gS[16][4*H_];
  __shared__ __align__(16) float zS[2][16][C_];
  int tid = threadIdx.x, wave = tid>>5, lane = tid&31;
  int row = lane&15, kg = (lane>>4)*2, rb = (lane>>4)*8;

  int colA = (wave*2)*16 + (lane&15);
  int colB = colA + 16;
  v2f bw0[12], bw1[12], bh0[24], bh1[24];
#pragma unroll
  for (int kk = 0; kk < 12; ++kk) {
    bw0[kk][0]=wih[colA*C_+kk*4+kg]; bw0[kk][1]=wih[colA*C_+kk*4+kg+1];
    bw1[kk][0]=wih[colB*C_+kk*4+kg]; bw1[kk][1]=wih[colB*C_+kk*4+kg+1];
  }
#pragma unroll
  for (int kk = 0; kk < 24; ++kk) {
    bh0[kk][0]=whh[colA*H_+kk*4+kg]; bh0[kk][1]=whh[colA*H_+kk*4+kg+1];
    bh1[kk][0]=whh[colB*H_+kk*4+kg]; bh1[kk][1]=whh[colB*H_+kk*4+kg+1];
  }
  float bs0 = bih[colA]+bhh[colA];
  float bs1 = bih[colB]+bhh[colB];
  v2f bl[24];
  float bls = 0.f;
  if (wave < 3) {                     // interlin weight fragments
    int colL = wave*16 + (lane&15);
#pragma unroll
    for (int kk = 0; kk < 24; ++kk) { bl[kk][0]=lw[colL*H_+kk*4+kg]; bl[kk][1]=lw[colL*H_+kk*4+kg+1]; }
    bls = lb[colL];
  }

  for (int i = tid; i < 16*H_; i += 384) {
    int r = i/H_, j = i%H_;
    int n = n0 + r;
    hS[r][j] = (n < BQ) ? h0[n*H_+j] : 0.f;
    cS[r][j] = (n < BQ) ? c0[n*H_+j] : 0.f;
  }
  if (tid < 192) {  // prefill t=0 tile (rows beyond BQ read scratch padding; results discarded)
    int r = tid/12, cc = (tid%12)*4;
    async_b128((unsigned)(size_t)&zS[0][r][cc], &z[((size_t)(n0+r)*T_ + 0)*C_ + cc]);
  }
  wait_async0();
  __syncthreads();

  for (int t = 0; t < T_; ++t) {
    int buf = t & 1;
    if (t+1 < T_ && tid < 192) {
      int r = tid/12, cc = (tid%12)*4;
      async_b128((unsigned)(size_t)&zS[buf^1][r][cc], &z[((size_t)(n0+r)*T_ + (t+1))*C_ + cc]);
    }
    v8f acc0, acc1;
#pragma unroll
    for (int r = 0; r < 8; ++r) { acc0[r]=bs0; acc1[r]=bs1; }
#pragma unroll
    for (int kk = 0; kk < 12; ++kk) {
      v2f a; a[0]=zS[buf][row][kk*4+kg]; a[1]=zS[buf][row][kk*4+kg+1];
      acc0 = wmma_f32(a, bw0[kk], acc0);
      acc1 = wmma_f32(a, bw1[kk], acc1);
    }
#pragma unroll
    for (int kk = 0; kk < 24; ++kk) {
      v2f a; a[0]=hS[row][kk*4+kg]; a[1]=hS[row][kk*4+kg+1];
      acc0 = wmma_f32(a, bh0[kk], acc0);
      acc1 = wmma_f32(a, bh1[kk], acc1);
    }
#pragma unroll
    for (int r = 0; r < 8; ++r) { gS[rb+r][colA] = acc0[r]; gS[rb+r][colB] = acc1[r]; }
    __syncthreads();
    for (int i = tid; i < 16*H_; i += 384) {
      int r=i/H_, j=i%H_;
      float ig=sigm(gS[r][j]), fg=sigm(gS[r][H_+j]);
      float gg=tanhf(gS[r][2*H_+j]), og=sigm(gS[r][3*H_+j]);
      float cc=fg*cS[r][j]+ig*gg;
      cS[r][j]=cc; hS[r][j]=og*tanhf(cc);
    }
    __syncthreads();
    if (wave < 3) {                   // interlin: h[16x96] @ lw^T[96x48]
      int colL = wave*16 + (lane&15);
      v8f acc;
#pragma unroll
      for (int r=0;r<8;++r) acc[r]=bls;
#pragma unroll
      for (int kk = 0; kk < 24; ++kk) {
        v2f a; a[0]=hS[row][kk*4+kg]; a[1]=hS[row][kk*4+kg+1];
        acc = wmma_f32(a, bl[kk], acc);
      }
#pragma unroll
      for (int r=0;r<8;++r){
        int n = n0 + rb + r;
        if (n < BQ) {
          int bb = n/QF, q = n%QF;
          int o = ((bb*T_ + t)*QF + q)*C_ + colL;
          inter[o] = acc[r] + intra[o];
        }
      }
    }
    wait_async0();
    __syncthreads();
  }
}

// ---------------- 6. generic WMMA GEMM: C = PReLU(A[MxK] * W[NxK]^T + bias) ----------------
__global__ void k_gemm(const float* __restrict__ A, const float* __restrict__ W,
                       const float* __restrict__ bias, const float* __restrict__ prelu_a,
                       float* __restrict__ C, int M, int N, int K)
{
  int wave = threadIdx.x >> 5, lane = threadIdx.x & 31;
  int tile = blockIdx.x * (blockDim.x >> 5) + wave;
  int tilesN = N >> 4;
  if (tile >= (M >> 4) * tilesN) return;
  int tm = tile / tilesN, tn = tile % tilesN;
  int row = lane & 15, kg = (lane>>4)*2, rb = (lane>>4)*8;
  int ar  = tm*16 + row;
  int col = tn*16 + (lane & 15);
  v8f acc;
  float bb = bias ? bias[col] : 0.f;
#pragma unroll
  for (int r=0;r<8;++r) acc[r]=bb;
  for (int k=0;k<K;k+=4){
    v2f a,b;
    a[0]=A[ar*K+k+kg];    a[1]=A[ar*K+k+kg+1];
    b[0]=W[col*K+k+kg];   b[1]=W[col*K+k+kg+1];
    acc = wmma_f32(a,b,acc);
  }
  float al = prelu_a ? prelu_a[0] : 0.f;
  bool pr = (prelu_a != nullptr);
#pragma unroll
  for (int r=0;r<8;++r){
    float v = acc[r];
    if (pr && v < 0.f) v *= al;
    C[(tm*16+rb+r)*N + col] = v;
  }
}

// ---------------- 7. per-head gather + LN ----------------
__global__ void k_head_ln(const float* __restrict__ P, int Pcols, int dsub,
                          const float* __restrict__ g, const float* __restrict__ beta,
                          float* __restrict__ dst, int rowOff, int rowsTot)
{
  int bh = blockIdx.x & 7, t = blockIdx.x >> 3;
  int b = bh >> 2, hh = bh & 3;
  int D = QF * dsub;
  __shared__ float r1[256], r2[256];
  __shared__ float mv[2];
  int tid = threadIdx.x;
  float s = 0.f, ss = 0.f;
  for (int d = tid; d < D; d += 256) {
    int q = d / dsub, e = d % dsub;
    float v = P[((b*T_ + t)*QF + q)*Pcols + hh*dsub + e];
    s += v; ss += v*v;
  }
  r1[tid]=s; r2[tid]=ss; __syncthreads();
  for (int st=128; st>0; st>>=1){ if(tid<st){ r1[tid]+=r1[tid+st]; r2[tid]+=r2[tid+st]; } __syncthreads(); }
  if (tid==0){ float m=r1[0]/(float)D; float v=r2[0]/(float)D - m*m; mv[0]=m; mv[1]=rsqrtf(v+EPSF); }
  __syncthreads();
  float m=mv[0], rs=mv[1];
  for (int d = tid; d < D; d += 256) {
    int q=d/dsub, e=d%dsub;
    float v = P[((b*T_+t)*QF+q)*Pcols + hh*dsub + e];
    dst[(bh*rowsTot + rowOff + t)*D + d] = (v-m)*rs*g[d] + beta[d];
  }
}

// ---------------- 7b. K/V ring-buffer copy into concat buffers ----------------
__global__ void k_copy_buf(const float* __restrict__ src, float* __restrict__ dst,
                           int rowsDst, int D, int total)
{
  int i = blockIdx.x*blockDim.x + threadIdx.x;
  if (i >= total) return;
  int per = LM1 * D;
  int bh = i / per, rem = i % per;
  dst[bh*rowsDst*D + rem] = src[i];
}

// ---------------- 8. local windowed attention ----------------
__global__ void k_attn(const float* __restrict__ Qm, const float* __restrict__ Kc,
                       const float* __restrict__ Vc, float* __restrict__ avm)
{
  int bh = blockIdx.x & 7, t = blockIdx.x >> 3;
  int b = bh >> 2, hh = bh & 3;
  __shared__ float qv[DQK];
  __shared__ float sc[L_];
  __shared__ float inv[1];
  int tid = threadIdx.x;
  for (int d = tid; d < DQK; d += 128) qv[d] = Qm[(bh*T_ + t)*DQK + d];
  // warm the V window in cache while scores compute (global_prefetch_b8)
  for (int l = tid; l < L_; l += 128)
    __builtin_prefetch(Vc + ((size_t)bh*(T_+LM1) + t + l)*DV, 0, 0);
  __syncthreads();
  const float scale = 0.14433756729740643f; // 1/sqrt(C)
  for (int l = tid; l < L_; l += 128) {
    const float* kr = Kc + ((size_t)bh*(T_+LM1) + t + l)*DQK;
    float s = 0.f;
    for (int d = 0; d < DQK; ++d) s += qv[d]*kr[d];
    sc[l] = s * scale;
  }
  __syncthreads();
  if (tid == 0) {
    float m = sc[0];
    for (int l = 1; l < L_; ++l) m = fmaxf(m, sc[l]);
    float su = 0.f;
    for (int l = 0; l < L_; ++l) { float e = __expf(sc[l]-m); sc[l]=e; su+=e; }
    inv[0] = 1.f/su;
  }
  __syncthreads();
  float iv = inv[0];
  for (int d = tid; d < DV; d += 128) {
    float s = 0.f;
    const float* vp = Vc + ((size_t)bh*(T_+LM1) + t)*DV + d;
    for (int l = 0; l < L_; ++l) s += sc[l] * vp[l*DV];
    s *= iv;
    int q = d / VD, vd = d % VD;
    avm[((b*T_+t)*QF + q)*C_ + hh*VD + vd] = s;
  }
}

// ---------------- 9. final LN over Qf*C + residual ----------------
__global__ void k_final_ln(const float* __restrict__ pout, const float* __restrict__ g,
                           const float* __restrict__ beta, const float* __restrict__ inter,
                           float* __restrict__ out)
{
  int bt = blockIdx.x;
  const float* p = pout + (size_t)bt*DP;
  __shared__ float r1[256], r2[256], mv[2];
  int tid = threadIdx.x;
  float s=0.f, ss=0.f;
  for (int d = tid; d < DP; d += 256){ float v = p[d]; s+=v; ss+=v*v; }
  r1[tid]=s; r2[tid]=ss; __syncthreads();
  for (int st=128; st>0; st>>=1){ if(tid<st){r1[tid]+=r1[tid+st]; r2[tid]+=r2[tid+st];} __syncthreads(); }
  if (tid==0){ float m=r1[0]/(float)DP; float v=r2[0]/(float)DP-m*m; mv[0]=m; mv[1]=rsqrtf(v+EPSF); }
  __syncthreads();
  float m=mv[0], rs=mv[1];
  for (int d=tid; d<DP; d+=256)
    out[(size_t)bt*DP + d] = (p[d]-m)*rs*g[d] + beta[d] + inter[(size_t)bt*DP + d];
}

extern "C" void kernel_launch(void* const* d_in, const int* in_sizes, int n_in,
                              void* d_out, int out_size, void* d_ws, size_t ws_size,
                              hipStream_t stream) {
  const float* x        = (const float*)d_in[0];
  const float* K_buf    = (const float*)d_in[1];
  const float* V_buf    = (const float*)d_in[2];
  const float* h0       = (const float*)d_in[3];
  const float* c0       = (const float*)d_in[4];
  const float* conv_w   = (const float*)d_in[5];
  const float* conv_b   = (const float*)d_in[6];
  const float* act_a    = (const float*)d_in[7];
  const float* norm_g   = (const float*)d_in[8];
  const float* norm_b   = (const float*)d_in[9];
  const float* wih_f    = (const float*)d_in[10];
  const float* whh_f    = (const float*)d_in[11];
  const float* bih_f    = (const float*)d_in[12];
  const float* bhh_f    = (const float*)d_in[13];
  const float* wih_b    = (const float*)d_in[14];
  const float* whh_b    = (const float*)d_in[15];
  const float* bih_b    = (const float*)d_in[16];
  const float* bhh_b    = (const float*)d_in[17];
  const float* deconv_w = (const float*)d_in[18];
  const float* deconv_b = (const float*)d_in[19];
  const float* inter_g  = (const float*)d_in[20];
  const float* inter_bb = (const float*)d_in[21];
  const float* iwih     = (const float*)d_in[22];
  const float* iwhh     = (const float*)d_in[23];
  const float* ibih     = (const float*)d_in[24];
  const float* ibhh     = (const float*)d_in[25];
  const float* lw       = (const float*)d_in[26];
  const float* lb       = (const float*)d_in[27];
  const float* q_w = (const float*)d_in[28]; const float* q_b  = (const float*)d_in[29];
  const float* q_a = (const float*)d_in[30]; const float* q_g  = (const float*)d_in[31];
  const float* q_be= (const float*)d_in[32];
  const float* k_w = (const float*)d_in[33]; const float* k_b  = (const float*)d_in[34];
  const float* k_a = (const float*)d_in[35]; const float* k_g  = (const float*)d_in[36];
  const float* k_be= (const float*)d_in[37];
  const float* v_w = (const float*)d_in[38]; const float* v_b  = (const float*)d_in[39];
  const float* v_a = (const float*)d_in[40]; const float* v_g  = (const float*)d_in[41];
  const float* v_be= (const float*)d_in[42];
  const float* p_w = (const float*)d_in[43]; const float* p_b  = (const float*)d_in[44];
  const float* p_a = (const float*)d_in[45]; const float* p_g  = (const float*)d_in[46];
  const float* p_be= (const float*)d_in[47];
  float* out = (float*)d_out;
  float* ws  = (float*)d_ws;

  float* u     = ws;
  float* y     = u     + (size_t)N_*QD*C_;        // 294912
  float* intra = y     + (size_t)N_*QD*2*H_;      // +1179648
  float* zln   = intra + (size_t)N_*DP;           // +1198080
  float* inter = zln   + (size_t)BQ*T_*C_;        // +1198080
  float* Qp    = inter + (size_t)N_*DP;           // +1198080
  float* Kp    = Qp    + (size_t)N_*QF*32;        // +798720
  float* Vp    = Kp    + (size_t)N_*QF*32;        // +798720
  float* Qm    = Vp    + (size_t)N_*QF*48;        // +1198080
  float* Kc    = Qm    + (size_t)BH*T_*DQK;       // +798720
  float* Vc    = Kc    + (size_t)BH*(T_+LM1)*DQK; // +1210560
  float* avm   = Vc    + (size_t)BH*(T_+LM1)*DV;  // +1815840
  float* pout  = avm   + (size_t)N_*DP;           // +1198080

  // intra path
  k_conv_ln<<<N_*QD, 64, 0, stream>>>(x, conv_w, conv_b, act_a, norm_g, norm_b, u);
  k_intra_lstm<<<48, 384, 0, stream>>>(u, wih_f, whh_f, bih_f, bhh_f,
                                       wih_b, whh_b, bih_b, bhh_b, y);
  k_deconv<<<N_*QF, 64, 0, stream>>>(y, deconv_w, deconv_b, x, intra);

  // inter path
  k_ln_inter<<<N_*QF, 64, 0, stream>>>(intra, inter_g, inter_bb, zln);
  k_inter_lstm<<<9, 384, 0, stream>>>(zln, iwih, iwhh, ibih, ibhh, h0, c0, lw, lb, intra, inter);

  // attention projections (M = B*T*Qf = 24960 rows)
  int M = N_*QF;
  k_gemm<<<(M/16)*(32/16)/4, 128, 0, stream>>>(inter, q_w, q_b, q_a, Qp, M, 32, C_);
  k_gemm<<<(M/16)*(32/16)/4, 128, 0, stream>>>(inter, k_w, k_b, k_a, Kp, M, 32, C_);
  k_gemm<<<(M/16)*(48/16)/4, 128, 0, stream>>>(inter, v_w, v_b, v_a, Vp, M, 48, C_);

  k_head_ln<<<BH*T_, 256, 0, stream>>>(Qp, 32, E_, q_g, q_be, Qm, 0,   T_);
  k_head_ln<<<BH*T_, 256, 0, stream>>>(Kp, 32, E_, k_g, k_be, Kc, LM1, T_+LM1);
  k_head_ln<<<BH*T_, 256, 0, stream>>>(Vp, 48, VD, v_g, v_be, Vc, LM1, T_+LM1);
  k_copy_buf<<<(BH*LM1*DQK + 255)/256, 256, 0, stream>>>(K_buf, Kc, T_+LM1, DQK, BH*LM1*DQK);
  k_copy_buf<<<(BH*LM1*DV  + 255)/256, 256, 0, stream>>>(V_buf, Vc, T_+LM1, DV,  BH*LM1*DV);

  k_attn<<<BH*T_, 128, 0, stream>>>(Qm, Kc, Vc, avm);

  // output projection + final LN + residual
  k_gemm<<<(M/16)*(48/16)/4, 128, 0, stream>>>(avm, p_w, p_b, p_a, pout, M, 48, C_);
  k_final_ln<<<N_, 256, 0, stream>>>(pout, p_g, p_be, inter, out);

  (void)in_sizes; (void)n_in; (void)out_size; (void)ws_size;
}